// MultiHeadAttention_16922171147101
// MI455X (gfx1250) — compile-verified
//
#include <hip/hip_runtime.h>
#include <hip/hip_bf16.h>
#include <math.h>

typedef float v2f __attribute__((ext_vector_type(2)));
typedef float v8f __attribute__((ext_vector_type(8)));

#define BATCH  256
#define SEQ    100
#define SEQP   112          // 7*16 padded
#define EMB    128
#define NH     8
#define KD     16
#define QSTR   132          // q staging row stride (floats), 4-bank skew per row
#define KSTR   17           // K/V row stride (floats), odd -> conflict-free
#define CSTR   116          // comp row stride over n
#define HSTR   18           // Q/head-out row stride
#define NTILES 7

// D = A(16x4 f32) * B(4x16 f32) + C(16x16 f32), full-precision CDNA5 WMMA
__device__ __forceinline__ v8f wmma4(v2f a, v2f b, v8f c) {
  return __builtin_amdgcn_wmma_f32_16x16x4_f32(false, a, false, b, (short)0, c,
                                               false, false);
}

__global__ __launch_bounds__(256, 1)
void mha_fused(const float* __restrict__ q, const float* __restrict__ embeds,
               const float* __restrict__ Wq, const float* __restrict__ Wk,
               const float* __restrict__ Wv, const float* __restrict__ Wo,
               const float* __restrict__ We, const float* __restrict__ Eb,
               float* __restrict__ out, float* __restrict__ outEmb)
{
  // lds_u: union of q staging [112][132] (14784) and comp [8][16][116] (14848)
  __shared__ float lds_u[14848];
  __shared__ float lds_K[NH * SEQP * KSTR];   // 15232 floats
  __shared__ float lds_V[NH * SEQP * KSTR];
  __shared__ float lds_Q[NH * 16 * HSTR];     // 2304 floats
  __shared__ float lds_hd[NH * 16 * HSTR];

  const int tid  = threadIdx.x;
  const int wave = tid >> 5;
  const int lane = tid & 31;
  const int l16  = lane & 15;
  const int lh   = lane >> 4;            // 0 or 1 (which half-wave)
  const int b    = blockIdx.y;
  const int q0   = blockIdx.x * 16;
  const int qv   = (SEQ - q0 < 16) ? (SEQ - q0) : 16;   // valid q rows in tile

  // ---------- phase 0: stream this block's embeds slab straight to the output
  // (reference returns embeds as 2nd output). Coalesced b128, warms L2 for e1.
  {
    const size_t base = ((size_t)b * SEQ + q0) * SEQ * (size_t)EMB;
    const float4* src = (const float4*)(embeds + base);
    float4*       dst = (float4*)(outEmb + base);
    const int n4 = qv * SEQ * EMB / 4;
    for (int i = tid; i < n4; i += 256) dst[i] = src[i];
  }

  // ---------- phase 1: stage q[b] into LDS, zero pad rows 100..111
  for (int i = tid; i < SEQP * EMB; i += 256) {
    const int n = i >> 7, c = i & 127;
    lds_u[n * QSTR + c] = (n < SEQ) ? q[((size_t)b * SEQ + n) * EMB + c] : 0.f;
  }
  __syncthreads();

  // ---------- phase 2: wave h computes head h's K,V (112 padded rows) + Q tile
  {
    const int h = wave;
    const float* WkB = Wk + (size_t)h * EMB * KD;
    const float* WvB = Wv + (size_t)h * EMB * KD;
    const float* WqB = Wq + (size_t)h * EMB * KD;
    for (int nt = 0; nt < NTILES; ++nt) {
      v8f ak = {0.f,0.f,0.f,0.f,0.f,0.f,0.f,0.f};
      v8f av = {0.f,0.f,0.f,0.f,0.f,0.f,0.f,0.f};
      for (int k0 = 0; k0 < EMB; k0 += 4) {
        const int i = k0 + lh * 2;
        v2f a = *(const v2f*)&lds_u[(nt * 16 + l16) * QSTR + i];   // A[M=n][K=i]
        v2f bk; bk.x = WkB[(i + 0) * KD + l16]; bk.y = WkB[(i + 1) * KD + l16];
        v2f bv; bv.x = WvB[(i + 0) * KD + l16]; bv.y = WvB[(i + 1) * KD + l16];
        ak = wmma4(a, bk, ak);
        av = wmma4(a, bv, av);
      }
      for (int j = 0; j < 8; ++j) {
        const int n = nt * 16 + j + lh * 8;
        lds_K[(h * SEQP + n) * KSTR + l16] = ak[j];
        lds_V[(h * SEQP + n) * KSTR + l16] = av[j];
      }
    }
    v8f aq = {0.f,0.f,0.f,0.f,0.f,0.f,0.f,0.f};
    for (int k0 = 0; k0 < EMB; k0 += 4) {
      const int i = k0 + lh * 2;
      v2f a = *(const v2f*)&lds_u[(q0 + l16) * QSTR + i];
      v2f bq; bq.x = WqB[(i + 0) * KD + l16]; bq.y = WqB[(i + 1) * KD + l16];
      aq = wmma4(a, bq, aq);
    }
    for (int j = 0; j < 8; ++j)
      lds_Q[(h * 16 + j + lh * 8) * HSTR + l16] = aq[j];
  }
  __syncthreads();   // lds_u is re-purposed as comp from here on

  // ---------- phase 3: comp[h][qi][n] = norm * Q·K^T   (norm = 1/sqrt(16))
  {
    const int h = wave;
    for (int nt = 0; nt < NTILES; ++nt) {
      v8f acc = {0.f,0.f,0.f,0.f,0.f,0.f,0.f,0.f};
      for (int k0 = 0; k0 < KD; k0 += 4) {
        const int i = k0 + lh * 2;
        v2f a = *(const v2f*)&lds_Q[(h * 16 + l16) * HSTR + i];    // A[qi][k]
        v2f bb;                                                    // B[k][n]=K[n][k]
        bb.x = lds_K[(h * SEQP + nt * 16 + l16) * KSTR + i + 0];
        bb.y = lds_K[(h * SEQP + nt * 16 + l16) * KSTR + i + 1];
        acc = wmma4(a, bb, acc);
      }
      for (int j = 0; j < 8; ++j)
        lds_u[(h * 16 + (j + lh * 8)) * CSTR + nt * 16 + l16] = 0.25f * acc[j];
    }
  }
  __syncthreads();

  // ---------- phase 4: edge bias  e1 = embeds·We^T (+b), tanh, add into comp.
  // 112 (qi,ntile) WMMA row-tiles split across waves; cols of C are heads.
  {
    const float bias = (l16 < NH) ? Eb[l16] : 0.f;
    for (int t = wave * 14; t < wave * 14 + 14; ++t) {
      const int qi = t / NTILES;
      const int nt = t % NTILES;
      const int qg = (q0 + qi < SEQ) ? (q0 + qi) : (SEQ - 1);      // clamp pads
      const int nr = nt * 16 + l16;
      const int ng = (nr < SEQ) ? nr : (SEQ - 1);
      const float* erow = embeds + (((size_t)b * SEQ + qg) * SEQ + ng) * EMB;
      v8f acc = {0.f,0.f,0.f,0.f,0.f,0.f,0.f,0.f};
      for (int k0 = 0; k0 < EMB; k0 += 4) {
        const int i = k0 + lh * 2;
        v2f a = *(const v2f*)(erow + i);                           // A[ni][e]
        v2f bb;                                                    // B[e][h]=We[h][e]
        if (l16 < NH) { bb.x = We[l16 * EMB + i]; bb.y = We[l16 * EMB + i + 1]; }
        else          { bb.x = 0.f;               bb.y = 0.f; }
        acc = wmma4(a, bb, acc);
      }
      if (l16 < NH) {
        for (int j = 0; j < 8; ++j) {
          const int ni = nt * 16 + j + lh * 8;
          lds_u[(l16 * 16 + qi) * CSTR + ni] += tanhf(acc[j] + bias);
        }
      }
    }
  }
  __syncthreads();

  // ---------- phase 5: softmax over n (exactly SEQ entries); zero pad cols
  {
    const int h = wave;
    for (int qi = 0; qi < 16; ++qi) {
      float* row = &lds_u[(h * 16 + qi) * CSTR];
      float v0 = row[lane], v1 = row[lane + 32], v2 = row[lane + 64];
      float v3 = (lane + 96 < SEQ) ? row[lane + 96] : -3.0e38f;
      float m = fmaxf(fmaxf(v0, v1), fmaxf(v2, v3));
      for (int off = 16; off; off >>= 1) m = fmaxf(m, __shfl_xor(m, off, 32));
      float e0 = __expf(v0 - m), e1 = __expf(v1 - m), e2 = __expf(v2 - m);
      float e3 = (lane + 96 < SEQ) ? __expf(v3 - m) : 0.f;
      float s = e0 + e1 + e2 + e3;
      for (int off = 16; off; off >>= 1) s += __shfl_xor(s, off, 32);
      const float r = 1.0f / s;
      row[lane]      = e0 * r;
      row[lane + 32] = e1 * r;
      row[lane + 64] = e2 * r;
      if (lane < 16) row[lane + 96] = e3 * r;   // n=96..111, zeros for n>=100
    }
  }
  __syncthreads();

  // ---------- phase 6: head output = attn(16x112, pads=0) x V(112x16, pads=0)
  {
    const int h = wave;
    v8f acc = {0.f,0.f,0.f,0.f,0.f,0.f,0.f,0.f};
    for (int k0 = 0; k0 < SEQP; k0 += 4) {
      const int i = k0 + lh * 2;
      v2f a = *(const v2f*)&lds_u[(h * 16 + l16) * CSTR + i];      // A[qi][n]
      v2f bb;                                                      // B[n][v]=V[n][v]
      bb.x = lds_V[(h * SEQP + i + 0) * KSTR + l16];
      bb.y = lds_V[(h * SEQP + i + 1) * KSTR + l16];
      acc = wmma4(a, bb, acc);
    }
    for (int j = 0; j < 8; ++j)
      lds_hd[(h * 16 + j + lh * 8) * HSTR + l16] = acc[j];
  }
  __syncthreads();

  // ---------- phase 7: out[b,q,:] = concat_heads(16x128) x W_out(128x128)
  // W_out flat (h*16+v, e) is already the row-major 128x128 B matrix.
  {
    const int ec = wave * 16;
    v8f acc = {0.f,0.f,0.f,0.f,0.f,0.f,0.f,0.f};
    for (int k0 = 0; k0 < EMB; k0 += 4) {
      const int hv = k0 + lh * 2;
      v2f a = *(const v2f*)&lds_hd[((hv >> 4) * 16 + l16) * HSTR + (hv & 15)];
      v2f bb;
      bb.x = Wo[(size_t)(hv + 0) * EMB + ec + l16];
      bb.y = Wo[(size_t)(hv + 1) * EMB + ec + l16];
      acc = wmma4(a, bb, acc);
    }
    for (int j = 0; j < 8; ++j) {
      const int qg = q0 + j + lh * 8;
      if (qg < SEQ)
        out[((size_t)b * SEQ + qg) * EMB + ec + l16] = acc[j];
    }
  }
}

extern "C" void kernel_launch(void* const* d_in, const int* in_sizes, int n_in,
                              void* d_out, int out_size, void* d_ws, size_t ws_size,
                              hipStream_t stream) {
  (void)in_sizes; (void)n_in; (void)out_size; (void)d_ws; (void)ws_size;
  const float* q  = (const float*)d_in[0];
  const float* em = (const float*)d_in[1];
  const float* Wq = (const float*)d_in[2];
  const float* Wk = (const float*)d_in[3];
  const float* Wv = (const float*)d_in[4];
  const float* Wo = (const float*)d_in[5];
  const float* We = (const float*)d_in[6];
  const float* Eb = (const float*)d_in[7];
  float* out    = (float*)d_out;
  float* outEmb = out + (size_t)BATCH * SEQ * EMB;   // (out, embeds) concat

  dim3 grid(NTILES, BATCH);
  mha_fused<<<grid, 256, 0, stream>>>(q, em, Wq, Wk, Wv, Wo, We, Eb, out, outEmb);
}